// Cosine_similarity_30142080483329
// MI455X (gfx1250) — compile-verified
//
#include <hip/hip_runtime.h>
#include <math.h>

typedef __attribute__((ext_vector_type(2))) float v2f;
typedef __attribute__((ext_vector_type(8))) float v8f;

#define FEAT 64
#define LSTR 68   // LDS row stride in floats (bank-conflict padding, 16B-aligned rows)

// Branch-free tanh: prefer the gfx1250 hardware transcendental.
__device__ __forceinline__ float fast_tanh(float x) {
#if __has_builtin(__builtin_amdgcn_tanhf)
    return __builtin_amdgcn_tanhf(x);
#elif __has_builtin(__builtin_amdgcn_tanh_f32)
    return __builtin_amdgcn_tanh_f32(x);
#else
    // branchless: tanh(|x|) = (1-e)/(1+e), e = exp(-2|x|); restore sign via bfi
    float ax = __builtin_fabsf(x);
    float e  = __expf(-2.0f * ax);
    float r  = __fdividef(1.0f - e, 1.0f + e);
    return __builtin_copysignf(r, x);
#endif
}

// ---------------- kernel 1: init first_idx ----------------
__global__ void k_init_first_idx(int* __restrict__ fi, const int* __restrict__ nu) {
    int u = blockIdx.x * 256 + threadIdx.x;
    if (u < *nu) fi[u] = 0x7FFFFFFF;
}

// ---------------- kernel 2: first occurrence via atomicMin ----------------
__global__ void k_first_occ(int* __restrict__ fi, const int* __restrict__ nodes, int n_ori) {
    int i = blockIdx.x * 256 + threadIdx.x;
    if (i < n_ori) atomicMin(&fi[nodes[i]], i);
}

// ---------------- kernel 3: fused gather + GEMM(tanh) + cosine fuse ----------------
// one wave (32 lanes) handles a 16-row block of unique nodes
__global__ void __launch_bounds__(256)
k_fused_proj_cos(const float* __restrict__ f1,
                 const float* __restrict__ f2,
                 const float* __restrict__ f3,
                 const float* __restrict__ trans,
                 const float* __restrict__ bias,
                 const int*  __restrict__ fi,
                 const int*  __restrict__ nu,
                 float* __restrict__ X,
                 int n_ori) {
    __shared__ __align__(16) float Bs[FEAT][LSTR];       // transposed: Bs[col][row] = trans[row][col]
    __shared__ __align__(16) float As[8][16][LSTR];      // per-wave gathered A tile (16 x 64)
    __shared__ int rowIdx[8][16];

    const int tid  = threadIdx.x;
    const int lane = tid & 31;
    const int wv   = tid >> 5;
    const int U    = *nu;

    // load trans transposed into LDS (whole workgroup, coalesced)
    #pragma unroll
    for (int j = 0; j < 16; ++j) {
        int idx = tid + 256 * j;          // 0..4095
        Bs[idx & 63][idx >> 6] = trans[idx];
    }
    __syncthreads();

    const int blk = blockIdx.x * 8 + wv;  // 16-row block id for this wave
    const int u0  = blk * 16;
    if (u0 >= U) return;                  // no barriers after this point

    const int half = (lane < 16) ? 0 : 1;
    const int l16  = lane & 15;
    const int koff = half ? 2 : 0;        // ISA A/B fragment layout: K split across half-waves

    // first-occurrence row indices for the 16 rows of this block
    if (lane < 16) {
        int rid = u0 + lane;
        int v = (rid < U) ? fi[rid] : 0;
        if (v >= n_ori || v < 0) v = 0;   // guaranteed present; clamp defensively
        rowIdx[wv][lane] = v;
    }
    __asm volatile("s_wait_dscnt 0" ::: "memory");

    float biasv[4];
    #pragma unroll
    for (int n = 0; n < 4; ++n) biasv[n] = bias[16 * n + l16];

    const float* fsrc[3] = { f1, f2, f3 };
    v8f h[3][4];                          // tanh(A@trans + bias), 3 matrices x 4 N-tiles

    #pragma unroll
    for (int m = 0; m < 3; ++m) {
        // stage 16 gathered rows (16 x 64 f32) into LDS; 256B coalesced per 16 lanes
        const float* __restrict__ F = fsrc[m];
        #pragma unroll
        for (int j = 0; j < 8; ++j) {
            int idx = j * 32 + lane;      // 0..255 float4 slots
            int row = idx >> 4;
            int c4  = idx & 15;
            const float4 v = *(const float4*)(F + (size_t)rowIdx[wv][row] * FEAT + c4 * 4);
            *(float4*)&As[wv][row][c4 * 4] = v;
        }
        __asm volatile("s_wait_dscnt 0" ::: "memory");

        // A fragments: lane<16 -> row l16, K=4k..4k+1 ; lane>=16 -> row l16, K=4k+2..4k+3
        v2f a[16];
        #pragma unroll
        for (int k = 0; k < 16; ++k)
            a[k] = *(const v2f*)&As[wv][l16][4 * k + koff];

        #pragma unroll
        for (int n = 0; n < 4; ++n) {
            v8f c = {};
            #pragma unroll
            for (int k = 0; k < 16; ++k) {
                // B fragment: b.x = trans[4k+koff][16n+l16], b.y = trans[4k+koff+1][16n+l16]
                v2f b = *(const v2f*)&Bs[16 * n + l16][4 * k + koff];
                c = __builtin_amdgcn_wmma_f32_16x16x4_f32(
                        false, a[k], false, b, (short)0, c, false, false);
            }
            #pragma unroll
            for (int r = 0; r < 8; ++r)
                c[r] = fast_tanh(c[r] + biasv[n]);
            h[m][n] = c;
        }
    }

    // ---- per-row dot products: C/D layout => lane holds col (16n+l16),
    // rows r..r+7 (lanes 0-15) or r+8..r+15 (lanes 16-31) across VGPR slots ----
    float d00[8], d01[8], d02[8], d11[8], d22[8];
    #pragma unroll
    for (int r = 0; r < 8; ++r) { d00[r] = d01[r] = d02[r] = d11[r] = d22[r] = 0.f; }
    #pragma unroll
    for (int n = 0; n < 4; ++n) {
        #pragma unroll
        for (int r = 0; r < 8; ++r) {
            float a0 = h[0][n][r], a1 = h[1][n][r], a2 = h[2][n][r];
            d00[r] += a0 * a0; d01[r] += a0 * a1; d02[r] += a0 * a2;
            d11[r] += a1 * a1; d22[r] += a2 * a2;
        }
    }
    // butterfly reduce over the 16 lanes holding the same row group
    #pragma unroll
    for (int off = 1; off < 16; off <<= 1) {
        #pragma unroll
        for (int r = 0; r < 8; ++r) {
            d00[r] += __shfl_xor(d00[r], off, 16);
            d01[r] += __shfl_xor(d01[r], off, 16);
            d02[r] += __shfl_xor(d02[r], off, 16);
            d11[r] += __shfl_xor(d11[r], off, 16);
            d22[r] += __shfl_xor(d22[r], off, 16);
        }
    }

    // cosine weights exactly per reference: w0=0, wn = w / (w1+w2)
    float wn1[8], wn2[8];
    #pragma unroll
    for (int r = 0; r < 8; ++r) {
        float n0 = __builtin_sqrtf(d00[r]);
        float n1 = __builtin_sqrtf(d11[r]);
        float n2 = __builtin_sqrtf(d22[r]);
        float w1 = __fdividef(d01[r], fmaxf(n0 * n1, 1e-8f));
        float w2 = __fdividef(d02[r], fmaxf(n0 * n2, 1e-8f));
        float inv_s = __frcp_rn(w1 + w2);
        wn1[r] = w1 * inv_s;
        wn2[r] = w2 * inv_s;
    }

    // X = h0 + wn1*h1 + wn2*h2, store to workspace
    #pragma unroll
    for (int n = 0; n < 4; ++n) {
        #pragma unroll
        for (int r = 0; r < 8; ++r) {
            int rowM = half ? (r + 8) : r;
            int urow = u0 + rowM;
            if (urow < U) {
                float x = h[0][n][r] + wn1[r] * h[1][n][r] + wn2[r] * h[2][n][r];
                X[(size_t)urow * FEAT + 16 * n + l16] = x;
            }
        }
    }
}

// ---------------- kernel 4: scatter X[nodes_ori] -> out ----------------
__global__ void k_scatter(const float4* __restrict__ X4,
                          const int* __restrict__ nodes,
                          float4* __restrict__ out4, int n_ori) {
    long t = (long)blockIdx.x * 256 + threadIdx.x;
    long total = (long)n_ori * 16;     // 16 float4 per 64-float row
    if (t < total) {
        int i = (int)(t >> 4);
        int c = (int)(t & 15);
        int row = nodes[i];
        out4[t] = X4[(size_t)row * 16 + c];
    }
}

extern "C" void kernel_launch(void* const* d_in, const int* in_sizes, int n_in,
                              void* d_out, int out_size, void* d_ws, size_t ws_size,
                              hipStream_t stream) {
    const float* f1    = (const float*)d_in[0];
    const float* f2    = (const float*)d_in[1];
    const float* f3    = (const float*)d_in[2];
    const float* trans = (const float*)d_in[3];
    const float* bias  = (const float*)d_in[4];
    const int*   nodes = (const int*)  d_in[5];
    const int*   nu    = (const int*)  d_in[6];
    const int n_ori = in_sizes[5];

    // workspace layout: [first_idx : n_ori ints][X : U*64 floats]
    int*   fi = (int*)d_ws;
    float* X  = (float*)((char*)d_ws + (size_t)n_ori * sizeof(int));

    int nb = (n_ori + 255) / 256;
    k_init_first_idx<<<nb, 256, 0, stream>>>(fi, nu);
    k_first_occ    <<<nb, 256, 0, stream>>>(fi, nodes, n_ori);

    int blocks16 = (n_ori + 15) / 16;          // upper bound on U/16 (U <= n_ori)
    int wgs = (blocks16 + 7) / 8;              // 8 waves (16-row blocks) per workgroup
    k_fused_proj_cos<<<wgs, 256, 0, stream>>>(f1, f2, f3, trans, bias, fi, nu, X, n_ori);

    long total = (long)n_ori * 16;
    int nb2 = (int)((total + 255) / 256);
    k_scatter<<<nb2, 256, 0, stream>>>((const float4*)X, nodes, (float4*)d_out, n_ori);
}